// UHGGraphSAGELayer_12524124635380
// MI455X (gfx1250) — compile-verified
//
#include <hip/hip_runtime.h>

// ---------------------------------------------------------------------------
// UHG GraphSAGE layer for MI455X (gfx1250, wave32).
//   K0 zero:   clear neigh_sum / neigh_w accumulators
//   K1 split:  x[50000,129] -> xf[50000,128] (16B aligned) + xh[50000]
//   K2 edge:   one wave32 per edge; coalesced float4 row gathers, ds_swizzle
//              butterfly reduction for the 3 UHG inner products, v_exp_f32,
//              global_atomic_add_f32 scatter (all resident in 192MB L2)
//   K3 prep:   inv_w = 1/max(neigh_w,eps); out[:,128] = relu(homog)
//   K4 gemm:   fp32 WMMA (v_wmma_f32_16x16x4_f32), B-stationary: each wave
//              keeps a 16-col slice of BOTH weight matrices in registers
//              (64 v2f frags) and streams 16-row A tiles; scale-by-inv_w fused
//              into A, ReLU fused into the store.
// ---------------------------------------------------------------------------

#define N_FEATS    128
#define ROW_STRIDE 129
#define KSTEPS     32          // 128 / 4
#define NTILES     8           // 128 / 16 output column tiles
#define WAVES_PER_NTILE 64

typedef __attribute__((ext_vector_type(2))) float v2f;
typedef __attribute__((ext_vector_type(8))) float v8f;

// wave32 butterfly add via ds_swizzle (group-of-32 xor mode, and_mask=0x1f)
#define SWZ_ADD(v, imm) \
  ((v) + __int_as_float(__builtin_amdgcn_ds_swizzle(__float_as_int(v), (imm))))

__device__ __forceinline__ float wave_reduce_add(float v) {
  v = SWZ_ADD(v, 0x041f);  // xor 1
  v = SWZ_ADD(v, 0x081f);  // xor 2
  v = SWZ_ADD(v, 0x101f);  // xor 4
  v = SWZ_ADD(v, 0x201f);  // xor 8
  v = SWZ_ADD(v, 0x401f);  // xor 16
  return v;                // all lanes hold the sum
}

__device__ __forceinline__ v8f wmma_f32(v2f a, v2f b, v8f c) {
  // 8-arg form: (neg_a, A, neg_b, B, c_mod, C, reuse_a, reuse_b)
  return __builtin_amdgcn_wmma_f32_16x16x4_f32(
      false, a, false, b, (short)0, c, false, false);
}

// --------------------------------------------------------------------- K0
__global__ void zero_kernel(float* __restrict__ p, size_t n) {
  size_t i = blockIdx.x * (size_t)blockDim.x + threadIdx.x;
  if (i < n) p[i] = 0.0f;
}

// --------------------------------------------------------------------- K1
__global__ void split_kernel(const float* __restrict__ x,
                             float* __restrict__ xf, float* __restrict__ xh,
                             int n_nodes) {
  size_t i = blockIdx.x * (size_t)blockDim.x + threadIdx.x;
  size_t total = (size_t)n_nodes * ROW_STRIDE;
  if (i >= total) return;
  size_t node = i / ROW_STRIDE;
  int    col  = (int)(i - node * ROW_STRIDE);
  float  v    = x[i];
  if (col < N_FEATS) xf[node * N_FEATS + col] = v;
  else               xh[node] = v;
}

// --------------------------------------------------------------------- K2
// edge_index is int64 per the reference setup (2 x n_edges, row-major).
__global__ void uhg_edge_kernel(const float* __restrict__ xf,
                                const float* __restrict__ xh,
                                const long long* __restrict__ erow,
                                const long long* __restrict__ ecol,
                                float* __restrict__ ns,   // [n,128] accum
                                float* __restrict__ nw,   // [n]    accum
                                int n_edges) {
  const int lane = threadIdx.x & 31;
  const int e = (int)((blockIdx.x * (size_t)blockDim.x + threadIdx.x) >> 5);
  if (e >= n_edges) return;

  const long long r = erow[e];
  const long long c = ecol[e];

  // coalesced 512B row gathers: lane l owns features 4l..4l+3
  const float4 av = ((const float4*)(xf + (size_t)r * N_FEATS))[lane];
  const float4 bv = ((const float4*)(xf + (size_t)c * N_FEATS))[lane];
  const float  ah = xh[r];
  const float  bh = xh[c];

  float paa = av.x * av.x + av.y * av.y + av.z * av.z + av.w * av.w;
  float pbb = bv.x * bv.x + bv.y * bv.y + bv.z * bv.z + bv.w * bv.w;
  float pab = av.x * bv.x + av.y * bv.y + av.z * bv.z + av.w * bv.w;

  paa = wave_reduce_add(paa);
  pbb = wave_reduce_add(pbb);
  pab = wave_reduce_add(pab);

  // UHG inner: -<feat,feat> + homog*homog
  const float aa  = ah * ah - paa;
  const float bb  = bh * bh - pbb;
  const float ab  = ah * bh - pab;
  const float den = aa * bb;
  const float sgn = (den > 0.0f) ? 1.0f : ((den < 0.0f) ? -1.0f : 0.0f);
  const float quad = (ab * ab - den) / (fmaxf(fabsf(den), 1e-9f) * sgn);
  const float w = __expf(-quad);

  float* dst = ns + (size_t)r * N_FEATS + lane * 4;
  atomicAdd(dst + 0, w * bv.x);
  atomicAdd(dst + 1, w * bv.y);
  atomicAdd(dst + 2, w * bv.z);
  atomicAdd(dst + 3, w * bv.w);
  if (lane == 0) atomicAdd(nw + r, w);
}

// --------------------------------------------------------------------- K3
__global__ void node_prep_kernel(const float* __restrict__ xh,
                                 const float* __restrict__ nw,
                                 float* __restrict__ invw,
                                 float* __restrict__ out, int n_nodes) {
  int i = blockIdx.x * blockDim.x + threadIdx.x;
  if (i >= n_nodes) return;
  invw[i] = 1.0f / fmaxf(nw[i], 1e-6f);
  out[(size_t)i * ROW_STRIDE + N_FEATS] = fmaxf(xh[i], 0.0f);  // relu(homog)
}

// --------------------------------------------------------------------- K4
// out[m, n] = relu( sum_k ns[m,k]*invw[m]*Wn[n,k] + xf[m,k]*Ws[n,k] )
// A frag (16x4 f32): lanes 0-15 -> M=lane, K={4s,4s+1}; lanes 16-31 -> K+2.
// B frag (4x16 f32): lanes 0-15 -> N=lane, K={4s,4s+1}; lanes 16-31 -> K+2.
// C/D (16x16 f32):   vgpr r, lanes 0-15 -> (M=r, N=lane); 16-31 -> (M=r+8).
__global__ void __launch_bounds__(256)
uhg_gemm_kernel(const float* __restrict__ xf, const float* __restrict__ ns,
                const float* __restrict__ invw,
                const float* __restrict__ Wn, const float* __restrict__ Ws,
                float* __restrict__ out, int n_nodes) {
  const int lane  = threadIdx.x & 31;
  const int gwave = (int)((blockIdx.x * (size_t)blockDim.x + threadIdx.x) >> 5);
  const int ntile = gwave / WAVES_PER_NTILE;          // 0..7
  const int mwave = gwave % WAVES_PER_NTILE;
  const int ncol  = ntile * 16 + (lane & 15);         // weight row = out col
  const int khalf = (lane >> 4) * 2;                  // 0 or 2

  // B-stationary: both weight slices live in registers for the whole kernel
  v2f bn[KSTEPS], bs[KSTEPS];
#pragma unroll
  for (int s = 0; s < KSTEPS; ++s) {
    const int kb = 4 * s + khalf;
    bn[s] = *(const v2f*)(Wn + (size_t)ncol * N_FEATS + kb);
    bs[s] = *(const v2f*)(Ws + (size_t)ncol * N_FEATS + kb);
  }

  const int n_mtiles = n_nodes >> 4;                  // 50000/16 = 3125 exact
  const int m_hi     = (lane >> 4) << 3;
  const int nout     = ntile * 16 + (lane & 15);

  for (int mt = mwave; mt < n_mtiles; mt += WAVES_PER_NTILE) {
    const int   mrow = mt * 16 + (lane & 15);
    const float iw   = invw[mrow];

    v8f acc = {};

    const float* arow = ns + (size_t)mrow * N_FEATS;
#pragma unroll
    for (int s = 0; s < KSTEPS; ++s) {
      v2f a = *(const v2f*)(arow + 4 * s + khalf);
      a.x *= iw;                                      // fused normalization
      a.y *= iw;
      acc = wmma_f32(a, bn[s], acc);
    }

    const float* frow = xf + (size_t)mrow * N_FEATS;
#pragma unroll
    for (int s = 0; s < KSTEPS; ++s) {
      v2f a = *(const v2f*)(frow + 4 * s + khalf);
      acc = wmma_f32(a, bs[s], acc);
    }

#pragma unroll
    for (int r = 0; r < 8; ++r) {                     // fused ReLU store
      const int m = mt * 16 + r + m_hi;
      out[(size_t)m * ROW_STRIDE + nout] = fmaxf(acc[r], 0.0f);
    }
  }
}

// ---------------------------------------------------------------------------
extern "C" void kernel_launch(void* const* d_in, const int* in_sizes, int n_in,
                              void* d_out, int out_size, void* d_ws,
                              size_t ws_size, hipStream_t stream) {
  const float*     x    = (const float*)d_in[0];
  const long long* eidx = (const long long*)d_in[1];   // int64 per reference
  const float*     Wn   = (const float*)d_in[2];
  const float*     Ws   = (const float*)d_in[3];
  float*           out  = (float*)d_out;

  const int n_nodes = in_sizes[0] / ROW_STRIDE;
  const int n_edges = in_sizes[1] / 2;
  const long long* erow = eidx;
  const long long* ecol = eidx + n_edges;

  // workspace layout (~52 MB)
  float* ns   = (float*)d_ws;                         // n*128 accum
  float* nw   = ns + (size_t)n_nodes * N_FEATS;       // n
  float* invw = nw + n_nodes;                         // n
  float* xf   = invw + n_nodes;                       // n*128 aligned feats
  float* xh   = xf + (size_t)n_nodes * N_FEATS;       // n homog

  {  // K0: clear accumulators (ns and nw are contiguous)
    size_t n = (size_t)n_nodes * N_FEATS + n_nodes;
    zero_kernel<<<dim3((unsigned)((n + 255) / 256)), dim3(256), 0, stream>>>(ns, n);
  }
  {  // K1: split x into aligned feats + homog
    size_t n = (size_t)n_nodes * ROW_STRIDE;
    split_kernel<<<dim3((unsigned)((n + 255) / 256)), dim3(256), 0, stream>>>(
        x, xf, xh, n_nodes);
  }
  {  // K2: one wave32 per edge (8 waves / 256-thread block)
    unsigned blocks = (unsigned)((n_edges + 7) / 8);
    uhg_edge_kernel<<<dim3(blocks), dim3(256), 0, stream>>>(
        xf, xh, erow, ecol, ns, nw, n_edges);
  }
  {  // K3: inv_w + homog column
    unsigned blocks = (unsigned)((n_nodes + 255) / 256);
    node_prep_kernel<<<dim3(blocks), dim3(256), 0, stream>>>(xh, nw, invw, out,
                                                             n_nodes);
  }
  {  // K4: fused dual-GEMM + ReLU via fp32 WMMA
    unsigned blocks = (NTILES * WAVES_PER_NTILE) / 8;  // 8 waves per block
    uhg_gemm_kernel<<<dim3(blocks), dim3(256), 0, stream>>>(xf, ns, invw, Wn,
                                                            Ws, out, n_nodes);
  }
}